// Memory_neural_27736898797576
// MI455X (gfx1250) — compile-verified
//
#include <hip/hip_runtime.h>
#include <math.h>

// Problem constants (from reference setup_inputs)
#define BATCH 32
#define SEQ   4096
#define EDIM  512
#define HDIM  1024

#define CHUNKS 16                          // S-chunks per batch for phase-B grid
#define ROWS_PER_BLOCK (SEQ / CHUNKS)      // 256
#define WAVES_PER_BLOCK 8
#define ROWS_PER_WAVE (ROWS_PER_BLOCK / WAVES_PER_BLOCK) // 32
#define PART_STRIDE (EDIM + 2)             // per-(b,chunk) partial: [m, l, acc[512]]

typedef float v2f __attribute__((ext_vector_type(2)));
typedef float v8f __attribute__((ext_vector_type(8)));

// ---------------------------------------------------------------------------
// Kernel 1: v_t[b,e] = tanh( sum_h Hd[b,h] * Ww[e,h] + Wb[e] )
// One wave per 16x16 output tile, V_WMMA_F32_16X16X4_F32 over K=HDIM.
//
// f32 WMMA fragment layout (ISA 7.12.2):
//   A (16x4  MxK): lanes 0-15 -> M=lane,  V0=K0, V1=K1 ; lanes 16-31 -> V0=K2, V1=K3
//   B (4x16  KxN): lanes 0-15 -> N=lane,  V0=K0, V1=K1 ; lanes 16-31 -> V0=K2, V1=K3
//   D (16x16):     VGPR r: lanes 0-15 -> M=r, N=lane ; lanes 16-31 -> M=r+8
// ---------------------------------------------------------------------------
__global__ __launch_bounds__(32) void vt_wmma_kernel(
    const float* __restrict__ Hd,   // [BATCH, HDIM]
    const float* __restrict__ Ww,   // [EDIM, HDIM]
    const float* __restrict__ Wb,   // [EDIM]
    float* __restrict__ vt)         // [BATCH, EDIM]
{
  const int tile  = blockIdx.x;      // 64 tiles: 2 (M) x 32 (N)
  const int m0    = (tile & 1) * 16;
  const int n0    = (tile >> 1) * 16;
  const int lane  = threadIdx.x & 31;
  const int half  = lane >> 4;       // 0: K pair {0,1}, 1: K pair {2,3}
  const int lan   = lane & 15;
  const int kOff  = half * 2;

  const float* arow = Hd + (size_t)(m0 + lan) * HDIM + kOff;  // A[m, k..k+1]
  const float* brow = Ww + (size_t)(n0 + lan) * HDIM + kOff;  // B[k..k+1, n] = Ww[n, k..k+1]

  v8f c = {};
#pragma unroll 8
  for (int k = 0; k < HDIM; k += 4) {
    v2f a = *(const v2f*)(arow + k);
    v2f b = *(const v2f*)(brow + k);
    // 8 args: (neg_a, A, neg_b, B, c_mod, C, reuse_a, reuse_b)
    c = __builtin_amdgcn_wmma_f32_16x16x4_f32(
        false, a, false, b, (short)0, c, false, false);
  }

  const float bias = Wb[n0 + lan];
#pragma unroll
  for (int r = 0; r < 8; ++r) {
    const int m = r + 8 * half;
    vt[(size_t)(m0 + m) * EDIM + (n0 + lan)] = tanhf(c[r] + bias);
  }
}

// ---------------------------------------------------------------------------
// Kernel 2: single-pass online-softmax + weighted embedding accumulation.
// Grid: (CHUNKS, BATCH) blocks of 256 threads (8 waves).
// Each wave owns 32 consecutive rows of S; a lane holds 16 E-elements as
// 4 x float4 at positions lane*4 + 128*j + {0..3} (coalesced 512B segments).
// Emits flash-style partials (m, l, acc[512]) per (batch, chunk).
// ---------------------------------------------------------------------------
__global__ __launch_bounds__(256) void attn_partial_kernel(
    const int*   __restrict__ mems,      // [BATCH, SEQ]
    const float* __restrict__ emb,       // [V, EDIM]
    const float* __restrict__ vt,        // [BATCH, EDIM]
    float* __restrict__ partials)        // [BATCH, CHUNKS, PART_STRIDE]
{
  const int b    = blockIdx.y;
  const int chnk = blockIdx.x;
  const int wave = threadIdx.x >> 5;
  const int lane = threadIdx.x & 31;

  // Per-lane v_t fragment (loaded once)
  float4 vf[4];
  const float4* vtp = (const float4*)(vt + (size_t)b * EDIM);
#pragma unroll
  for (int j = 0; j < 4; ++j) vf[j] = vtp[lane + 32 * j];

  // Coalesced index load: one int per lane, broadcast later via __shfl.
  const int sBase = chnk * ROWS_PER_BLOCK + wave * ROWS_PER_WAVE;
  const int idxv  = mems[(size_t)b * SEQ + sBase + lane];

  float m = -3.0e38f;
  float l = 0.0f;
  float4 acc[4];
#pragma unroll
  for (int j = 0; j < 4; ++j) acc[j] = make_float4(0.f, 0.f, 0.f, 0.f);

  for (int i = 0; i < ROWS_PER_WAVE; ++i) {
    const int idx = __shfl(idxv, i, 32);
    const float4* row = (const float4*)(emb + (size_t)idx * EDIM);

    // Prefetch next gathered row (global_prefetch_b8); lanes span the 2KB row.
    if (i + 1 < ROWS_PER_WAVE) {
      const int idxn = __shfl(idxv, i + 1, 32);
      __builtin_prefetch(emb + (size_t)idxn * EDIM + lane * 16, 0, 1);
    }

    float4 ef[4];
#pragma unroll
    for (int j = 0; j < 4; ++j) ef[j] = row[lane + 32 * j];

    // Partial dot, then full wave32 reduction.
    float d = 0.0f;
#pragma unroll
    for (int j = 0; j < 4; ++j) {
      d += ef[j].x * vf[j].x + ef[j].y * vf[j].y +
           ef[j].z * vf[j].z + ef[j].w * vf[j].w;
    }
#pragma unroll
    for (int off = 16; off > 0; off >>= 1) d += __shfl_xor(d, off, 32);

    // Online softmax update (all lanes hold identical d, m, l).
    const float mNew  = fmaxf(m, d);
    const float scale = __expf(m - mNew);   // 0 on first iteration (underflow)
    const float p     = __expf(d - mNew);
    l = l * scale + p;
#pragma unroll
    for (int j = 0; j < 4; ++j) {
      acc[j].x = acc[j].x * scale + p * ef[j].x;
      acc[j].y = acc[j].y * scale + p * ef[j].y;
      acc[j].z = acc[j].z * scale + p * ef[j].z;
      acc[j].w = acc[j].w * scale + p * ef[j].w;
    }
    m = mNew;
  }

  // ---- merge the 8 waves of this block (conflict-free, no atomics) ----
  __shared__ float lds_acc[WAVES_PER_BLOCK][EDIM];   // 16 KB
  __shared__ float lds_m[WAVES_PER_BLOCK];
  __shared__ float lds_l[WAVES_PER_BLOCK];

  float4* ldsv = (float4*)lds_acc[wave];
#pragma unroll
  for (int j = 0; j < 4; ++j) ldsv[lane + 32 * j] = acc[j];
  if (lane == 0) { lds_m[wave] = m; lds_l[wave] = l; }
  __syncthreads();

  float M = -3.0e38f;
#pragma unroll
  for (int w = 0; w < WAVES_PER_BLOCK; ++w) M = fmaxf(M, lds_m[w]);
  float scw[WAVES_PER_BLOCK];
  float L = 0.0f;
#pragma unroll
  for (int w = 0; w < WAVES_PER_BLOCK; ++w) {
    scw[w] = __expf(lds_m[w] - M);
    L += lds_l[w] * scw[w];
  }

  float* p = partials + ((size_t)b * CHUNKS + chnk) * PART_STRIDE;
  if (threadIdx.x == 0) { p[0] = M; p[1] = L; }
#pragma unroll
  for (int e = threadIdx.x; e < EDIM; e += 256) {
    float a = 0.0f;
#pragma unroll
    for (int w = 0; w < WAVES_PER_BLOCK; ++w) a += lds_acc[w][e] * scw[w];
    p[2 + e] = a;
  }
}

// ---------------------------------------------------------------------------
// Kernel 3: combine the 16 chunk partials per batch and normalize.
// out[b, 0, e] = (sum_c acc_c[e] * exp(m_c - M)) / (sum_c l_c * exp(m_c - M))
// ---------------------------------------------------------------------------
__global__ __launch_bounds__(256) void attn_final_kernel(
    const float* __restrict__ partials,  // [BATCH, CHUNKS, PART_STRIDE]
    float* __restrict__ out)             // [BATCH, EDIM]
{
  const int b = blockIdx.x;
  const float* pb = partials + (size_t)b * CHUNKS * PART_STRIDE;

  float M = -3.0e38f;
#pragma unroll
  for (int c = 0; c < CHUNKS; ++c) M = fmaxf(M, pb[c * PART_STRIDE]);

  float sc[CHUNKS];
  float L = 0.0f;
#pragma unroll
  for (int c = 0; c < CHUNKS; ++c) {
    sc[c] = __expf(pb[c * PART_STRIDE] - M);
    L += pb[c * PART_STRIDE + 1] * sc[c];
  }
  const float inv = 1.0f / L;

  for (int e = threadIdx.x; e < EDIM; e += 256) {
    float a = 0.0f;
#pragma unroll
    for (int c = 0; c < CHUNKS; ++c) a += pb[c * PART_STRIDE + 2 + e] * sc[c];
    out[(size_t)b * EDIM + e] = a * inv;
  }
}

// ---------------------------------------------------------------------------
extern "C" void kernel_launch(void* const* d_in, const int* in_sizes, int n_in,
                              void* d_out, int out_size, void* d_ws, size_t ws_size,
                              hipStream_t stream) {
  (void)in_sizes; (void)n_in; (void)out_size; (void)ws_size;

  const float* Hd   = (const float*)d_in[0];   // [32, 1024]
  const int*   mems = (const int*)  d_in[1];   // [32, 4096]
  const float* emb  = (const float*)d_in[2];   // [50257, 512]
  const float* Ww   = (const float*)d_in[3];   // [512, 1024]
  const float* Wb   = (const float*)d_in[4];   // [512]
  float* out = (float*)d_out;                  // [32, 1, 512]

  float* vt       = (float*)d_ws;                      // 32*512 floats
  float* partials = vt + (size_t)BATCH * EDIM;         // 32*16*514 floats

  vt_wmma_kernel<<<64, 32, 0, stream>>>(Hd, Ww, Wb, vt);

  dim3 grid(CHUNKS, BATCH);
  attn_partial_kernel<<<grid, 256, 0, stream>>>(mems, emb, vt, partials);

  attn_final_kernel<<<BATCH, 256, 0, stream>>>(partials, out);
}